// RGCNPredictor_9517647528308
// MI455X (gfx1250) — compile-verified
//
#include <hip/hip_runtime.h>

// RGCN (2 layers) for MI455X / gfx1250: wave32, WMMA f32 16x16x4, TDM
// (tensor_load_to_lds) double-buffered LDS staging for both GEMM operands.
//
// Reformulation: per-(dst,relation) MEAN aggregation of *input* features first
// (cheap atomic scatter), then one dense GEMM per layer:
//   out = acc[N, R*F] @ Wflat[R*F, H] + x @ root + bias
// Layer1 GEMM: [20000x512]x[512x128], Layer2 GEMM: [20000x4096]x[4096x128].

#define RELS 32
#define EMB  16
#define HID  128
#define KS   64     // K-slab staged per TDM transfer

typedef __attribute__((ext_vector_type(2))) float    v2f;
typedef __attribute__((ext_vector_type(8))) float    v8f;
typedef __attribute__((ext_vector_type(4))) unsigned v4u;
typedef __attribute__((ext_vector_type(8))) unsigned v8u;

__device__ __forceinline__ void atomAddF32(float* p, float v) {
  // relaxed / agent scope -> hardware global_atomic_add_f32 (no return)
  __hip_atomic_fetch_add(p, v, __ATOMIC_RELAXED, __HIP_MEMORY_SCOPE_AGENT);
}

// low 32 bits of a flat LDS pointer == byte offset within the block's LDS
__device__ __forceinline__ unsigned lds_off(const void* p) {
  return (unsigned)(uintptr_t)p;
}

// ---- Tensor Data Mover descriptor packing (CDNA5 ISA) --
// Group 1 (256b): [15:0]=wg_mask(0) [17:16]=data_size [20]=pad_en
// [24:22]=pad_interval [31:25]=pad_amount [79:48]=tensor_dim0
// [111:80]=tensor_dim1 [127:112]=tile_dim0 [143:128]=tile_dim1
// [159:144]=tile_dim2(0) [207:160]=dim0_stride [255:208]=dim1_stride(0 for 2D)
__device__ __forceinline__ v8u tdm_group1(unsigned td0, unsigned td1,
                                          unsigned tile0, unsigned tile1,
                                          unsigned long long stride0,
                                          unsigned pad_en, unsigned pad_ivl,
                                          unsigned pad_amt) {
  v8u g;
  g[0] = (2u << 16) | (pad_en << 20) | (pad_ivl << 22) | (pad_amt << 25);
  g[1] = (td0 & 0xffffu) << 16;
  g[2] = ((td0 >> 16) & 0xffffu) | ((td1 & 0xffffu) << 16);
  g[3] = ((td1 >> 16) & 0xffffu) | ((tile0 & 0xffffu) << 16);
  g[4] = tile1 & 0xffffu;                       // tile_dim2 = 0
  g[5] = (unsigned)stride0;
  g[6] = (unsigned)((stride0 >> 32) & 0xffffu); // dim1_stride = 0 (2D tile)
  g[7] = 0u;
  return g;
}

// Group 0 (128b): [1:0]=count(1) [63:32]=lds_addr [120:64]=global_addr
// [127:126]=type(2).  Groups 2/3 zero: tile_dim3/4 = 0 (2D tile).
__device__ __forceinline__ void tdm_load_2d(unsigned lds_addr,
                                            unsigned long long gaddr, v8u g1) {
  v4u g0;
  g0.x = 1u;
  g0.y = lds_addr;
  g0.z = (unsigned)gaddr;
  g0.w = ((unsigned)(gaddr >> 32) & 0x01ffffffu) | (2u << 30);
  v4u gz = {0u, 0u, 0u, 0u};
  asm volatile("tensor_load_to_lds %0, %1, %2, %3"
               :: "s"(g0), "s"(g1), "s"(gz), "s"(gz)
               : "memory");
}

// ---------------------------------------------------------------------------
__global__ void zero_f32_kernel(float* __restrict__ p, size_t n) {
  size_t i = (size_t)blockIdx.x * blockDim.x + threadIdx.x;
  size_t stride = (size_t)gridDim.x * blockDim.x;
  for (; i < n; i += stride) p[i] = 0.0f;
}

__global__ void count_kernel(const int* __restrict__ eidx, const int* __restrict__ etype,
                             float* __restrict__ counts, int E) {
  int e = blockIdx.x * blockDim.x + threadIdx.x;
  if (e >= E) return;
  int dst = eidx[E + e];
  int r   = etype[e];
  atomAddF32(&counts[dst * RELS + r], 1.0f);
}

__global__ void inv_kernel(const float* __restrict__ counts, float* __restrict__ inv, int n) {
  int i = blockIdx.x * blockDim.x + threadIdx.x;
  if (i >= n) return;
  float c = counts[i];
  inv[i] = (c > 0.0f) ? (1.0f / c) : 0.0f;
}

// acc[dst][rel][f] += x[src][f] * inv[dst,rel]
template <int F>
__global__ void scatter_kernel(const int* __restrict__ eidx, const int* __restrict__ etype,
                               const float* __restrict__ x, const float* __restrict__ inv,
                               float* __restrict__ acc, int E) {
  long long t = (long long)blockIdx.x * blockDim.x + threadIdx.x;
  int e = (int)(t / F);
  int f = (int)(t % F);
  if (e >= E) return;
  int src = eidx[e];
  int dst = eidx[E + e];
  int r   = etype[e];
  float w = inv[dst * RELS + r];
  atomAddF32(&acc[((long long)dst * RELS + r) * F + f],
             x[(long long)src * F + f] * w);
}

// ---------------------------------------------------------------------------
// One 256-thread block (8 waves) per 16-row M-tile; wave w owns columns
// [16w, 16w+16).  A (16xKS) and W (KSx128) slabs staged in LDS by the TDM,
// double-buffered: DMA of slab s+1 overlaps WMMA compute on slab s.
// A-slab uses TDM LDS padding (1 dword per 64) -> row stride 65 floats, so
// the 16-row A-fragment column reads are bank-conflict free.
template <bool RELU>
__global__ __launch_bounds__(256) void
rgcn_gemm_wmma(const float* __restrict__ A, int K,
               const float* __restrict__ W,
               const float* __restrict__ X, int Kr,
               const float* __restrict__ Rt,
               const float* __restrict__ bias,
               float* __restrict__ out) {
  __shared__ float sA[2][16][KS + 1];   // TDM-padded rows
  __shared__ float sW[2][KS][HID];

  const int tid  = threadIdx.x;
  const int wave = tid >> 5;
  const int lane = tid & 31;
  const int half = lane >> 4;          // 0: K pair {0,1}, 1: K pair {2,3}
  const int l16  = lane & 15;
  const int m0   = blockIdx.x * 16;
  const int col  = wave * 16 + l16;
  const int nslab = K / KS;

  const unsigned long long aBase =
      (unsigned long long)(uintptr_t)A + (unsigned long long)m0 * K * 4ull;
  const unsigned long long wBase = (unsigned long long)(uintptr_t)W;

  // A: tensor [16 x K] starting at row m0; tile 16 x KS; pad 1 dword / 64
  const v8u g1A = tdm_group1((unsigned)K, 16u, KS, 16u,
                             (unsigned long long)K, 1u, 5u, 0u);
  // W: tensor [K x 128]; tile KS x 128; no padding
  const v8u g1W = tdm_group1(HID, (unsigned)K, HID, KS,
                             (unsigned long long)HID, 0u, 0u, 0u);

  if (wave == 0) {                     // TDM ignores EXEC: issue once per WG
    tdm_load_2d(lds_off(&sA[0][0][0]), aBase, g1A);
    tdm_load_2d(lds_off(&sW[0][0][0]), wBase, g1W);
    tdm_load_2d(lds_off(&sA[1][0][0]), aBase + (unsigned long long)KS * 4, g1A);
    tdm_load_2d(lds_off(&sW[1][0][0]), wBase + (unsigned long long)KS * HID * 4, g1W);
  }

  v8f c = {};

  for (int s = 0; s < nslab; ++s) {
    const int buf = s & 1;
    if (wave == 0) {
      if (s + 1 < nslab) __builtin_amdgcn_s_wait_tensorcnt(2);
      else               __builtin_amdgcn_s_wait_tensorcnt(0);
    }
    __syncthreads();                    // slab s visible to all 8 waves

#pragma unroll
    for (int kk = 0; kk < KS; kk += 4) {
      const int ka = kk + 2 * half;
      v2f a;
      a.x = sA[buf][l16][ka];
      a.y = sA[buf][l16][ka + 1];
      v2f b;
      b.x = sW[buf][ka][col];
      b.y = sW[buf][ka + 1][col];
      c = __builtin_amdgcn_wmma_f32_16x16x4_f32(false, a, false, b,
                                                (short)0, c, false, false);
    }

    __syncthreads();                    // all waves done reading buf
    if (wave == 0 && s + 2 < nslab) {
      const unsigned long long ks2 = (unsigned long long)(s + 2) * KS;
      tdm_load_2d(lds_off(&sA[buf][0][0]), aBase + ks2 * 4, g1A);
      tdm_load_2d(lds_off(&sW[buf][0][0]), wBase + ks2 * HID * 4, g1W);
    }
  }

  // root (self-loop) term, small K: straight from global/L2
  const float* xrow = X + (size_t)(m0 + l16) * Kr;
#pragma unroll 4
  for (int k0 = 0; k0 < Kr; k0 += 4) {
    const int ka = k0 + 2 * half;
    v2f a = *(const v2f*)(xrow + ka);   // 8B aligned: Kr even, ka even
    v2f b;
    b.x = Rt[(size_t)(ka + 0) * HID + col];
    b.y = Rt[(size_t)(ka + 1) * HID + col];
    c = __builtin_amdgcn_wmma_f32_16x16x4_f32(false, a, false, b,
                                              (short)0, c, false, false);
  }

  const float bv = bias[col];
#pragma unroll
  for (int i = 0; i < 8; ++i) {         // C/D: row = m0 + i + 8*half, col
    float v = c[i] + bv;
    if (RELU) v = fmaxf(v, 0.0f);
    out[(size_t)(m0 + i + 8 * half) * HID + col] = v;
  }
}

// ---------------------------------------------------------------------------
extern "C" void kernel_launch(void* const* d_in, const int* in_sizes, int n_in,
                              void* d_out, int out_size, void* d_ws, size_t ws_size,
                              hipStream_t stream) {
  const int*   eidx  = (const int*)d_in[0];    // [2, E]
  const int*   etype = (const int*)d_in[1];    // [E]
  const float* x     = (const float*)d_in[2];  // [N, 16]
  const float* W1    = (const float*)d_in[3];  // [32,16,128] == flat [512,128]
  const float* root1 = (const float*)d_in[4];  // [16,128]
  const float* b1    = (const float*)d_in[5];  // [128]
  const float* W2    = (const float*)d_in[6];  // [32,128,128] == flat [4096,128]
  const float* root2 = (const float*)d_in[7];  // [128,128]
  const float* b2    = (const float*)d_in[8];  // [128]
  float* out = (float*)d_out;

  const int E = in_sizes[1];
  const int N = in_sizes[2] / EMB;             // 20000, divisible by 16

  // workspace carve-out (floats)
  float* ws = (float*)d_ws;
  size_t off = 0;
  float* counts = ws + off; off += (size_t)N * RELS;
  float* inv    = ws + off; off += (size_t)N * RELS;
  float* acc1   = ws + off; off += (size_t)N * RELS * EMB;
  float* h1     = ws + off; off += (size_t)N * HID;
  float* acc2   = ws + off; off += (size_t)N * RELS * HID;
  (void)ws_size;

  zero_f32_kernel<<<4096, 256, 0, stream>>>(ws, off);

  count_kernel<<<(E + 255) / 256, 256, 0, stream>>>(eidx, etype, counts, E);
  inv_kernel<<<((N * RELS) + 255) / 256, 256, 0, stream>>>(counts, inv, N * RELS);

  dim3 gtile(N / 16);                    // 1250 blocks x 8 waves

  // ---- layer 1 ----
  {
    long long work = (long long)E * EMB;
    scatter_kernel<EMB><<<(unsigned)((work + 255) / 256), 256, 0, stream>>>(
        eidx, etype, x, inv, acc1, E);
    rgcn_gemm_wmma<true><<<gtile, 256, 0, stream>>>(
        acc1, RELS * EMB, W1, x, EMB, root1, b1, h1);
  }

  // ---- layer 2 ----
  {
    long long work = (long long)E * HID;
    scatter_kernel<HID><<<(unsigned)((work + 255) / 256), 256, 0, stream>>>(
        eidx, etype, h1, inv, acc2, E);
    rgcn_gemm_wmma<false><<<gtile, 256, 0, stream>>>(
        acc2, RELS * HID, W2, h1, HID, root2, b2, out);
  }
}